// FasterRCNNSofterLabels_43198781063711
// MI455X (gfx1250) — compile-verified
//
#include <hip/hip_runtime.h>
#include <hip/hip_bf16.h>

#define G_GT 64
#define K1   8                 // anchors per thread, pass 1
#define K2   4                 // anchors per thread, pass 2
#define THREADS 256

typedef int v4i __attribute__((ext_vector_type(4)));
// Address-space-qualified pointer typedefs (attributes bind to pointee here).
typedef __attribute__((address_space(1))) v4i* as1_v4i_ptr;
typedef __attribute__((address_space(3))) v4i* as3_v4i_ptr;

#if defined(__HIP_DEVICE_COMPILE__) && __has_builtin(__builtin_amdgcn_global_load_async_to_lds_b128)
#define HAVE_ASYNC_LDS 1
#else
#define HAVE_ASYNC_LDS 0
#endif

// Stage 16 bytes from global into LDS (per-lane). Async on gfx1250, plain copy otherwise.
__device__ __forceinline__ void stage16(const void* gsrc, void* ldst) {
#if HAVE_ASYNC_LDS
  __builtin_amdgcn_global_load_async_to_lds_b128(
      (as1_v4i_ptr)(gsrc), (as3_v4i_ptr)(ldst), /*imm offset*/0, /*cpol*/0);
#else
  *(float4*)ldst = *(const float4*)gsrc;
#endif
}

__device__ __forceinline__ void wait_async_lds() {
#if HAVE_ASYNC_LDS
#if __has_builtin(__builtin_amdgcn_s_wait_asynccnt)
  __builtin_amdgcn_s_wait_asynccnt(0);
#else
  asm volatile("s_wait_asynccnt 0" ::: "memory");
#endif
#endif
}

// IoU used by BOTH passes so the bit patterns match exactly (needed for the
// mq == highest_per_gt equality test in the matcher's low-quality restore).
__device__ __forceinline__ float iou_box(float ax0, float ay0, float ax1, float ay1,
                                         float bx0, float by0, float bx1, float by1) {
  float area_a = (ax1 - ax0) * (ay1 - ay0);
  float area_b = (bx1 - bx0) * (by1 - by0);
  float ltx = fmaxf(ax0, bx0), lty = fmaxf(ay0, by0);
  float rbx = fminf(ax1, bx1), rby = fminf(ay1, by1);
  float w = fmaxf(rbx - ltx, 0.0f), h = fmaxf(rby - lty, 0.0f);
  float inter = w * h;
  return inter / (area_a + area_b - inter);
}

__global__ void frcnn_init_kernel(unsigned int* __restrict__ hi_bits, int n) {
  int i = blockIdx.x * blockDim.x + threadIdx.x;
  if (i < n) hi_bits[i] = 0u;   // 0-bits == 0.0f, identity for max of non-neg IoU
}

// Pass 1: highest IoU per GT over all anchors (bit-pattern atomic max).
__global__ void frcnn_pass1_kernel(const float* __restrict__ anchors,
                                   const float* __restrict__ gts,
                                   unsigned int* __restrict__ hi_bits, int A) {
  const int b = blockIdx.y;
  const int tid = threadIdx.x;
  __shared__ float sgt[G_GT * 4];

  if (tid < G_GT)
    stage16(&gts[((size_t)b * G_GT + tid) * 4], &sgt[tid * 4]);
  wait_async_lds();
  __syncthreads();

  float vmax[G_GT];
#pragma unroll
  for (int g = 0; g < G_GT; ++g) vmax[g] = 0.0f;

  const float4* ab = (const float4*)(anchors + (size_t)b * A * 4);
  const int base = blockIdx.x * (THREADS * K1);
#pragma unroll 1
  for (int k = 0; k < K1; ++k) {
    int a = base + tid + k * THREADS;
    if (a < A) {
      if (a + THREADS < A) __builtin_prefetch(&ab[a + THREADS], 0, 0);  // global_prefetch_b8
      float4 box = ab[a];
#pragma unroll
      for (int g = 0; g < G_GT; ++g) {
        float iou = iou_box(sgt[4 * g], sgt[4 * g + 1], sgt[4 * g + 2], sgt[4 * g + 3],
                            box.x, box.y, box.z, box.w);
        vmax[g] = fmaxf(vmax[g], iou);
      }
    }
  }

  unsigned int* out = hi_bits + b * G_GT;
  const int lane = tid & 31;  // wave32
#pragma unroll
  for (int g = 0; g < G_GT; ++g) {
    float v = vmax[g];
#pragma unroll
    for (int off = 16; off > 0; off >>= 1)
      v = fmaxf(v, __shfl_xor(v, off, 32));
    if (lane == 0) atomicMax(&out[g], __float_as_uint(v));  // monotone: IoU >= 0
  }
}

// Pass 2: matcher + label logic + matched-box gather.
__global__ void frcnn_pass2_kernel(const float* __restrict__ anchors,
                                   const float* __restrict__ gts,
                                   const float* __restrict__ scores,
                                   const int* __restrict__ confs,
                                   const unsigned int* __restrict__ hi_bits,
                                   float* __restrict__ labels_out,
                                   float* __restrict__ boxes_out, int A) {
  const int b = blockIdx.y;
  const int tid = threadIdx.x;
  __shared__ float sgt[G_GT * 4];
  __shared__ float sscore[G_GT];
  __shared__ int   sconf[G_GT];
  __shared__ float shi[G_GT];     // raw bit patterns from pass 1 == float IoU values

  if (tid < 64) {
    stage16(&gts[((size_t)b * G_GT + tid) * 4], &sgt[tid * 4]);
  } else if (tid < 80) {
    int i = tid - 64;
    stage16(&scores[b * G_GT + i * 4], &sscore[i * 4]);
  } else if (tid < 96) {
    int i = tid - 80;
    stage16(&confs[b * G_GT + i * 4], &sconf[i * 4]);
  } else if (tid < 112) {
    int i = tid - 96;
    stage16(&hi_bits[b * G_GT + i * 4], &shi[i * 4]);
  }
  wait_async_lds();
  __syncthreads();

  const float4* ab = (const float4*)(anchors + (size_t)b * A * 4);
  float*  lab = labels_out + (size_t)b * A;
  float4* mgb = (float4*)boxes_out + (size_t)b * A;

  const int base = blockIdx.x * (THREADS * K2);
#pragma unroll 1
  for (int k = 0; k < K2; ++k) {
    int a = base + tid + k * THREADS;
    if (a >= A) continue;
    float4 box = ab[a];

    float best = -1.0f;
    int bg = 0;
    bool lowq = false;
#pragma unroll
    for (int g = 0; g < G_GT; ++g) {
      float iou = iou_box(sgt[4 * g], sgt[4 * g + 1], sgt[4 * g + 2], sgt[4 * g + 3],
                          box.x, box.y, box.z, box.w);
      if (iou > best) { best = iou; bg = g; }   // strict '>' => first-index argmax (JAX)
      lowq |= (iou == shi[g]);                  // low-quality restore
    }

    int midx;
    if (lowq)             midx = bg;
    else if (best < 0.3f) midx = -1;   // BELOW_LOW
    else if (best < 0.7f) midx = -2;   // BETWEEN
    else                  midx = bg;

    const int  cl  = midx >= 0 ? midx : 0;
    const bool pos = midx >= 0;
    const float sc = sscore[cl];

    float label = pos ? 1.0f : 0.0f;
    label = fminf(label, sc);
    if (midx == -1)          label = 0.0f;
    if (midx == -2)          label = -1.0f;
    if (sc < 1.0f && pos)    label = -1.0f;
    if (sconf[cl] == 0 && pos) label = -1.0f;

    lab[a] = label;
    mgb[a] = make_float4(sgt[4 * cl], sgt[4 * cl + 1], sgt[4 * cl + 2], sgt[4 * cl + 3]);
  }
}

extern "C" void kernel_launch(void* const* d_in, const int* in_sizes, int n_in,
                              void* d_out, int out_size, void* d_ws, size_t ws_size,
                              hipStream_t stream) {
  const float* anchors = (const float*)d_in[0];  // [B, A, 4]
  const float* gts     = (const float*)d_in[1];  // [B, 64, 4]
  const float* scores  = (const float*)d_in[2];  // [B, 64]
  const int*   confs   = (const int*)d_in[3];    // [B, 64]

  const int B = in_sizes[2] / G_GT;
  const int A = in_sizes[0] / (4 * B);

  unsigned int* hi_bits = (unsigned int*)d_ws;      // B*64 words
  float* labels_out = (float*)d_out;                // [B, A]
  float* boxes_out  = (float*)d_out + (size_t)B * A; // [B, A, 4]

  const int nhi = B * G_GT;
  frcnn_init_kernel<<<(nhi + THREADS - 1) / THREADS, THREADS, 0, stream>>>(hi_bits, nhi);

  dim3 g1((A + THREADS * K1 - 1) / (THREADS * K1), B);
  frcnn_pass1_kernel<<<g1, THREADS, 0, stream>>>(anchors, gts, hi_bits, A);

  dim3 g2((A + THREADS * K2 - 1) / (THREADS * K2), B);
  frcnn_pass2_kernel<<<g2, THREADS, 0, stream>>>(anchors, gts, scores, confs, hi_bits,
                                                 labels_out, boxes_out, A);
}